// LengthRegulator_6305011990987
// MI455X (gfx1250) — compile-verified
//
#include <hip/hip_runtime.h>
#include <stdint.h>

#define BATCH 32
#define LEN 512
#define DIM 384
#define MAXT 4096
#define ROW_BYTES (DIM * 4)       /* 1536 bytes per frame */
#define IN_ROWS 16                /* input rows staged per block */

// ---------------------------------------------------------------------------
// Kernel 1: per-batch inclusive scan of durations (512 elements, one block).
// Writes cs[] to workspace and total (as float) to the tail of d_out.
// ---------------------------------------------------------------------------
__global__ void lr_scan_kernel(const int* __restrict__ duration,
                               int* __restrict__ cs,
                               float* __restrict__ out_total) {
  __shared__ int s[LEN];
  const int b = blockIdx.x;
  const int tid = threadIdx.x;
  int v = duration[b * LEN + tid];
  if (v < 0) v = 0;
  s[tid] = v;
  __syncthreads();
#pragma unroll
  for (int off = 1; off < LEN; off <<= 1) {
    int add = (tid >= off) ? s[tid - off] : 0;
    __syncthreads();
    s[tid] += add;
    __syncthreads();
  }
  cs[b * LEN + tid] = s[tid];
  if (tid == LEN - 1) out_total[b] = (float)s[tid];
}

// ---------------------------------------------------------------------------
// Kernel 2: scatter expansion through LDS using the gfx1250 async path.
// Block = 256 threads (8 waves). Each wave owns 2 input rows:
//   stage each row into LDS via global_load_async_to_lds_b128 (3 x 512B
//   chunks, NT hint: each x row is read exactly once), s_wait_asynccnt 0
//   (async LDS accesses are mutually unordered), then replicate the row to
//   out[t] for t in [cs[i-1], min(cs[i], MAXT)) via
//   global_store_async_from_lds_b128 (NT hint: 201 MB streaming write vs
//   192 MB L2 -> avoid thrash).
// Each wave touches only its own LDS rows -> no workgroup barrier needed.
// ---------------------------------------------------------------------------
__global__ void lr_expand_kernel(const float* __restrict__ x,
                                 const int* __restrict__ cs,
                                 float* __restrict__ out) {
  __shared__ __align__(16) char smem[IN_ROWS * ROW_BYTES];  // 24 KB
  const int b = blockIdx.y;
  const int i0 = blockIdx.x * IN_ROWS;
  const int wave = threadIdx.x >> 5;   // 0..7
  const int lane = threadIdx.x & 31;

  // LDS byte offset of our staging buffer (flat LDS aperture keeps the
  // allocation offset in the low 32 bits of the generic pointer).
  const unsigned lds_base = (unsigned)(uintptr_t)(void*)smem;
  const int* csb = cs + b * LEN;

  // ---- Phase 1: async-stage 2 input rows per wave into LDS ----
#pragma unroll
  for (int r = 0; r < 2; ++r) {
    const int i = i0 + wave * 2 + r;
    const unsigned lrow = lds_base + (unsigned)((wave * 2 + r) * ROW_BYTES);
    const unsigned long long gsrc =
        (unsigned long long)(const void*)(x + ((size_t)b * LEN + i) * DIM);
#pragma unroll
    for (int c = 0; c < 3; ++c) {
      const unsigned ldst = lrow + (unsigned)(c * 512 + lane * 16);
      const unsigned long long ga = gsrc + (unsigned long long)(c * 512 + lane * 16);
      asm volatile("global_load_async_to_lds_b128 %0, %1, off th:TH_LOAD_NT"
                   :: "v"(ldst), "v"(ga)
                   : "memory");
    }
  }
  // Drain the async loads before any store-from-LDS may read the staged data
  // (ASYNCcnt is per-wave; async LDS accesses are unordered among themselves).
  asm volatile("s_wait_asynccnt 0x0" ::: "memory");

  // ---- Phase 2: replicate each staged row to its output slots ----
#pragma unroll
  for (int r = 0; r < 2; ++r) {
    const int i = i0 + wave * 2 + r;
    const int start = (i == 0) ? 0 : csb[i - 1];
    int end = csb[i];
    if (end > MAXT) end = MAXT;
    if (start >= end) continue;       // zero-duration row or fully clipped
    const unsigned lrow = lds_base + (unsigned)((wave * 2 + r) * ROW_BYTES);
    for (int t = start; t < end; ++t) {
      const unsigned long long gdst =
          (unsigned long long)(void*)(out + ((size_t)b * MAXT + t) * DIM);
#pragma unroll
      for (int c = 0; c < 3; ++c) {
        const unsigned lsrc = lrow + (unsigned)(c * 512 + lane * 16);
        const unsigned long long ga = gdst + (unsigned long long)(c * 512 + lane * 16);
        asm volatile("global_store_async_from_lds_b128 %0, %1, off th:TH_STORE_NT"
                     :: "v"(ga), "v"(lsrc)
                     : "memory");
      }
    }
  }
  asm volatile("s_wait_asynccnt 0x0" ::: "memory");
}

// ---------------------------------------------------------------------------
// Kernel 3: zero the invalid tail rows t in [min(total, MAXT), MAXT).
// ---------------------------------------------------------------------------
__global__ void lr_tail_zero_kernel(const int* __restrict__ cs,
                                    float* __restrict__ out) {
  const int b = blockIdx.y;
  int total = cs[b * LEN + (LEN - 1)];
  if (total > MAXT) total = MAXT;

  const size_t e = (size_t)blockIdx.x * blockDim.x + threadIdx.x;  // elem in batch
  // Whole-block early out: last element of this block still in a valid row?
  const size_t blk_last = ((size_t)blockIdx.x + 1) * blockDim.x - 1;
  if ((int)(blk_last / DIM) < total) return;

  const int t = (int)(e / DIM);
  if (t >= total) out[(size_t)b * MAXT * DIM + e] = 0.0f;
}

// ---------------------------------------------------------------------------
extern "C" void kernel_launch(void* const* d_in, const int* in_sizes, int n_in,
                              void* d_out, int out_size, void* d_ws, size_t ws_size,
                              hipStream_t stream) {
  const float* x        = (const float*)d_in[0];
  const int*   duration = (const int*)d_in[1];
  // d_in[2] = max_len scalar (fixed at 4096 by the reference setup)

  float* out       = (float*)d_out;
  float* out_total = out + (size_t)BATCH * MAXT * DIM;  // 32 floats after the tensor
  int*   cs        = (int*)d_ws;                        // 32*512 ints of scratch

  // 1) per-batch inclusive cumsum + totals
  lr_scan_kernel<<<dim3(BATCH), dim3(LEN), 0, stream>>>(duration, cs, out_total);

  // 2) async-LDS scatter expansion: grid = (512/16 row-chunks, 32 batches)
  lr_expand_kernel<<<dim3(LEN / IN_ROWS, BATCH), dim3(256), 0, stream>>>(x, cs, out);

  // 3) zero the invalid tail
  const int tail_blocks = (MAXT * DIM) / 256;  // 6144
  lr_tail_zero_kernel<<<dim3(tail_blocks, BATCH), dim3(256), 0, stream>>>(cs, out);
}